// Sampler_89386859364711
// MI455X (gfx1250) — compile-verified
//
#include <hip/hip_runtime.h>
#include <hip/hip_bf16.h>
#include <stdint.h>

#define B_ROWS      256
#define VOCAB       128000
#define TILE_F4     256            // one float4 per thread per tile
#define GREEDY_EPS  1e-6f

struct Part { float sv; int si; float gv; int gi; };

// ---- CDNA5 async global->LDS copy (gfx1250): one 16B chunk per lane ----
__device__ __forceinline__ void async_ld_b128(unsigned lds_off, const void* g) {
  asm volatile("global_load_async_to_lds_b128 %0, %1, off"
               :: "v"(lds_off), "v"((unsigned long long)(uintptr_t)g)
               : "memory");
}
// Wait until <= N async ops outstanding (in-order completion => oldest retire first).
#define WAIT_ASYNC(n) asm volatile("s_wait_asynccnt %0" :: "i"(n) : "memory")
// All of this wave's LDS reads done (WAR guard before async engine reuses a buffer).
#define WAIT_DS0()    asm volatile("s_wait_dscnt 0" ::: "memory")

__device__ __forceinline__ void take_better(float& v, int& i, float ov, int oi) {
  if (ov > v || (ov == v && oi < i)) { v = ov; i = oi; }
}

template<int NSEG>
__global__ __launch_bounds__(256)
void sampler_partial(const float* __restrict__ logits,
                     const float* __restrict__ temps,
                     const float* __restrict__ u,
                     Part* __restrict__ parts)
{
  constexpr int NITER = VOCAB / (NSEG * TILE_F4 * 4);  // tiles per segment
  constexpr int NBUF  = 4;
  constexpr int AHEAD = NBUF - 1;                      // tiles issued ahead
  static_assert(NITER > AHEAD, "pipeline needs NITER > AHEAD");

  __shared__ float4 bl[NBUF][TILE_F4];                 // 16 KB logits ring
  __shared__ float4 bu[NBUF][TILE_F4];                 // 16 KB u ring
  __shared__ float s_sv[8]; __shared__ int s_si[8];
  __shared__ float s_gv[8]; __shared__ int s_gi[8];

  const int t   = threadIdx.x;
  const int seg = blockIdx.x;
  const int row = blockIdx.y;
  constexpr int seg_f4 = NITER * TILE_F4;              // float4s per segment

  const float temp  = temps[row];
  const float inv_t = (temp <= GREEDY_EPS) ? 1.0f : (1.0f / temp);

  const float4* lrow = (const float4*)(logits + (size_t)row * VOCAB) + (size_t)seg * seg_f4;
  const float4* urow = (const float4*)(u      + (size_t)row * VOCAB) + (size_t)seg * seg_f4;
  const int idx_base0 = seg * seg_f4 * 4;              // segment's first element index

  float sv = -__builtin_inff(); int si = 0x7fffffff;   // sampled argmax
  float gv = -__builtin_inff(); int gi = 0x7fffffff;   // greedy argmax

  auto issue = [&](int tile) {
    const int b = tile & (NBUF - 1);
    async_ld_b128((unsigned)(uintptr_t)&bl[b][t], lrow + tile * TILE_F4 + t);
    async_ld_b128((unsigned)(uintptr_t)&bu[b][t], urow + tile * TILE_F4 + t);
  };
  auto consume = [&](int tile) {
    const int b = tile & (NBUF - 1);
    const float4 lv = bl[b][t];                        // read exactly what we loaded
    const float4 uv = bu[b][t];
    const int base = idx_base0 + (tile * TILE_F4 + t) * 4;
#define STEP(C, J) {                                          \
      const float l = lv.C;                                   \
      const float s = fmaf(l, inv_t, -__logf(-__logf(uv.C))); \
      const int idx = base + (J);                             \
      if (s > sv) { sv = s; si = idx; }                       \
      if (l > gv) { gv = l; gi = idx; } }
    STEP(x,0) STEP(y,1) STEP(z,2) STEP(w,3)
#undef STEP
  };

  // Prologue: fill the ring AHEAD tiles deep.
  issue(0); issue(1); issue(2);

  // Steady state: 3 pairs (3 KB/wave) in flight after each wait.
#pragma unroll 1
  for (int it = 0; it < NITER - AHEAD; ++it) {
    WAIT_DS0();                 // prior tile's LDS reads done before ring reuse
    issue(it + AHEAD);          // 8 async ops outstanding
    WAIT_ASYNC(2 * AHEAD);      // retire the pair for tile `it`
    consume(it);
  }
  // Drain.
  WAIT_ASYNC(4); consume(NITER - 3);
  WAIT_ASYNC(2); consume(NITER - 2);
  WAIT_ASYNC(0); consume(NITER - 1);

  // wave32 butterfly reduction (gfx1250 is wave32-only)
  for (int off = 16; off > 0; off >>= 1) {
    float osv = __shfl_xor(sv, off, 32); int osi = __shfl_xor(si, off, 32);
    take_better(sv, si, osv, osi);
    float ogv = __shfl_xor(gv, off, 32); int ogi = __shfl_xor(gi, off, 32);
    take_better(gv, gi, ogv, ogi);
  }
  const int wave = t >> 5;
  if ((t & 31) == 0) { s_sv[wave]=sv; s_si[wave]=si; s_gv[wave]=gv; s_gi[wave]=gi; }
  __syncthreads();
  if (t == 0) {
    for (int w = 1; w < 8; ++w) {
      take_better(sv, si, s_sv[w], s_si[w]);
      take_better(gv, gi, s_gv[w], s_gi[w]);
    }
    Part p; p.sv = sv; p.si = si; p.gv = gv; p.gi = gi;
    parts[(size_t)row * NSEG + seg] = p;
  }
}

__global__ __launch_bounds__(256)
void sampler_final(const Part* __restrict__ parts,
                   const float* __restrict__ temps,
                   float* __restrict__ out, int nseg)
{
  const int r = threadIdx.x;                 // 256 threads, one per row
  float sv = -__builtin_inff(); int si = 0x7fffffff;
  float gv = -__builtin_inff(); int gi = 0x7fffffff;
  for (int s = 0; s < nseg; ++s) {
    const Part p = parts[(size_t)r * nseg + s];
    take_better(sv, si, p.sv, p.si);
    take_better(gv, gi, p.gv, p.gi);
  }
  const int choice = (temps[r] <= GREEDY_EPS) ? gi : si;
  out[r] = (float)choice;                    // index < 2^24: exact in f32
}

extern "C" void kernel_launch(void* const* d_in, const int* in_sizes, int n_in,
                              void* d_out, int out_size, void* d_ws, size_t ws_size,
                              hipStream_t stream) {
  const float* logits = (const float*)d_in[0];
  const float* temps  = (const float*)d_in[1];
  const float* u      = (const float*)d_in[2];
  float* out = (float*)d_out;
  Part* parts = (Part*)d_ws;

  // 25 segments/row (6400 blocks) if workspace allows; degrade to 5 or 1.
  if (ws_size >= (size_t)B_ROWS * 25 * sizeof(Part)) {
    sampler_partial<25><<<dim3(25, B_ROWS), 256, 0, stream>>>(logits, temps, u, parts);
    sampler_final<<<1, 256, 0, stream>>>(parts, temps, out, 25);
  } else if (ws_size >= (size_t)B_ROWS * 5 * sizeof(Part)) {
    sampler_partial<5><<<dim3(5, B_ROWS), 256, 0, stream>>>(logits, temps, u, parts);
    sampler_final<<<1, 256, 0, stream>>>(parts, temps, out, 5);
  } else {
    sampler_partial<1><<<dim3(1, B_ROWS), 256, 0, stream>>>(logits, temps, u, parts);
    sampler_final<<<1, 256, 0, stream>>>(parts, temps, out, 1);
  }
}